// ComplexUpSampling2D_78821239816764
// MI455X (gfx1250) — compile-verified
//
#include <hip/hip_runtime.h>

typedef unsigned int u32;
typedef unsigned long long u64;
typedef __attribute__((ext_vector_type(4))) u32   v4u;
typedef __attribute__((ext_vector_type(8))) int   v8i;
typedef __attribute__((ext_vector_type(4))) int   v4i;
typedef __attribute__((ext_vector_type(4))) float fvec4;   // raw vector (nontemporal-legal)

#define BB 16
#define HH 128
#define WW 128
#define CC 64
#define WCHUNK 16      // input w positions per block (16 threads per w: 64ch/4)
#define THREADS 256

// LDS tile: one output-row segment, w already duplicated:
// 32 w_out x 64 ch x complex64 = 16 KiB
#define TILE_FLOATS (WCHUNK * 2 * CC * 2)

__global__ __launch_bounds__(THREADS)
void ComplexUpSampling2D_kernel(const float* __restrict__ xre,
                                const float* __restrict__ xim,
                                float* __restrict__ out)
{
    __shared__ float tile[TILE_FLOATS];

    const u32 t     = threadIdx.x;
    const u32 blk   = blockIdx.x;
    const u32 chunk = blk & 7;            // WW / WCHUNK = 8 chunks per row
    const u32 h     = (blk >> 3) & (HH - 1);
    const u32 b     = blk >> 10;          // blk / (8*128)
    const u32 w0    = chunk * WCHUNK;

    // ---- load: fully contiguous 4 KiB per plane per block -----------------
    const u64 in_base = (((u64)(b * HH + h)) * WW + w0) * CC;   // float elems
    const fvec4 re = *(const fvec4*)(xre + in_base + (u64)t * 4);
    const fvec4 im = *(const fvec4*)(xim + in_base + (u64)t * 4);

    const u32 w_local = t >> 4;           // 0..15
    const u32 c       = (t & 15) * 4;     // channel (complex index)

    // interleave into complex pairs
    const fvec4 p0 = { re.x, im.x, re.y, im.y };
    const fvec4 p1 = { re.z, im.z, re.w, im.w };

    // LDS layout: [w_out_local(32)][ch(64)][2 floats] — contiguous like global
    float* d0 = &tile[((2 * w_local)     * CC + c) * 2];
    float* d1 = &tile[((2 * w_local + 1) * CC + c) * 2];
    *(fvec4*)(d0)     = p0;  *(fvec4*)(d0 + 4) = p1;
    *(fvec4*)(d1)     = p0;  *(fvec4*)(d1 + 4) = p1;

    __syncthreads();

    // ---- store: TDM replays the tile to out rows 2h and 2h+1 --------------
#if defined(__HIP_DEVICE_COMPILE__) && __has_builtin(__builtin_amdgcn_tensor_store_from_lds)
    if (t == 0) {
        const u64 gaddr = (u64)(uintptr_t)out +
            ((((u64)(b * 2 * HH + 2 * h)) * (2 * WW) + 2 * w0) * CC) * 8ull;
        const u32 lds_off = (u32)(uintptr_t)(&tile[0]); // addr[31:0] = LDS byte offset

        // D# group0: count=1 valid, lds_addr, global_addr[56:0], type=2
        v4u g0 = { 1u,
                   lds_off,
                   (u32)gaddr,
                   (u32)((gaddr >> 32) & 0x01FFFFFFu) | 0x80000000u };

        // D# group1:
        //  [17:16] data_size=3 (8B)  [19] iterate_enable=1
        //  tensor_dim0/1 huge (never OOB), tile_dim0 = 2048 (x8B = 16 KiB tile)
        const u32 tdim  = 0x40000000u;
        const u32 tile0 = WCHUNK * 2 * CC;          // 2048 complex elements
        v8i g1;
        g1[0] = (int)((3u << 16) | (1u << 19));     // data_size=8B, iterate
        g1[1] = (int)((tdim & 0xFFFFu) << 16);      // tensor_dim0[15:0]
        g1[2] = (int)(((tdim >> 16) & 0xFFFFu) |    // tensor_dim0[31:16]
                      ((tdim & 0xFFFFu) << 16));    // tensor_dim1[15:0]
        g1[3] = (int)(((tdim >> 16) & 0xFFFFu) |    // tensor_dim1[31:16]
                      (tile0 << 16));               // tile_dim0
        g1[4] = 0;                                  // tile_dim1=0, tile_dim2=0
        g1[5] = (int)tile0;                         // tensor_dim0_stride[31:0]
        g1[6] = 0;                                  // stride0 hi / stride1 lo
        g1[7] = 0;                                  // stride1 hi

        // D# group2 (iterate mode):
        //  tensor_dim2 big; lds_addr_increment=0 (replay same tile);
        //  global_addr_increment = one output row = 2*WW*CC (8B units);
        //  iterate_count=1 -> 2 iterations (rows 2h and 2h+1)
        v4i g2 = { (int)0x40000000, 0, (int)(2 * WW * CC), (int)(1u << 16) };
        v4i g3 = { 0, 0, 0, 0 };

        // 6-arg form (clang-23 / therock-10.0 lane): extra int32x8 slot zeroed,
        // cache-policy last.
        v8i gx = { 0, 0, 0, 0, 0, 0, 0, 0 };
        __builtin_amdgcn_tensor_store_from_lds(g0, g1, g2, g3, gx, 0);
        // no explicit s_wait_tensorcnt: S_ENDPGM's implicit wait-idle holds the
        // workgroup (and its LDS) until the DMA drains; maximizes overlap.
    }
#else
    // Fallback: cooperative nontemporal copy LDS -> global, both output rows.
    const u64 row0       = ((((u64)(b * 2 * HH + 2 * h)) * (2 * WW) + 2 * w0) * CC) * 2ull;
    const u64 row_stride = (u64)(2 * WW) * CC * 2;   // floats per out row
    for (int r = 0; r < 2; ++r) {
        float* dst = out + row0 + (u64)r * row_stride;
        #pragma unroll
        for (int i = 0; i < TILE_FLOATS / (THREADS * 4); ++i) {
            const u32 off = (i * THREADS + t) * 4;
            fvec4 v = *(fvec4*)&tile[off];
            __builtin_nontemporal_store(v, (fvec4*)(dst + off));
        }
    }
#endif
}

extern "C" void kernel_launch(void* const* d_in, const int* in_sizes, int n_in,
                              void* d_out, int out_size, void* d_ws, size_t ws_size,
                              hipStream_t stream) {
    (void)in_sizes; (void)n_in; (void)out_size; (void)d_ws; (void)ws_size;
    const float* xre = (const float*)d_in[0];
    const float* xim = (const float*)d_in[1];
    float* out = (float*)d_out;   // complex64 viewed as interleaved float pairs

    const int blocks = BB * HH * (WW / WCHUNK);   // 16 * 128 * 8 = 16384
    ComplexUpSampling2D_kernel<<<blocks, THREADS, 0, stream>>>(xre, xim, out);
}